// ForwardXRayVolumeRenderer_82643760709640
// MI455X (gfx1250) — compile-verified
//
#include <hip/hip_runtime.h>

// ---------------------------------------------------------------------------
// X-ray volume renderer for MI455X (gfx1250, wave32).
//
// Gather/latency-bound: 65,536 rays x 320 samples, 8-tap trilinear from a
// 256^3 f32 volume (64 MB, ~L2-resident at 192 MB). Compute is ~0.9 GFLOP
// total -> matrix cores are irrelevant; the chip features that matter are the
// async global->LDS DMA path, prefetch, and wave32 occupancy. Two code paths:
//
//  FAST (taken when R[1][0]==0 && R[2][0]==0, e.g. identity rotation):
//    one block == one image row -> fy/fz are block-uniform per sample, so the
//    8 taps live on 4 volume x-lines differing only in x. The block stages
//    those 4 rows (4 KB) into LDS with ONE global_load_async_to_lds_b128 per
//    wave per sample (8 waves x 32 lanes x 16 B), double-buffered so the DMA
//    for sample s+1 overlaps the trilinear math of sample s. Sync via
//    s_wait_asynccnt + workgroup barrier. Taps then read from LDS.
//
//  GENERIC: branch-free masked per-lane gathers with global_prefetch_b8
//    issued ahead along the ray.
//
// Max-reduction: wave32 shuffles + atomicMax on float bits (values >= 0, so
// unsigned bit compare == float max). Normalize runs in-place on d_out.
// ---------------------------------------------------------------------------

namespace {
constexpr int   IMG_W = 256;
constexpr int   IMG_H = 256;
constexpr int   N_PTS = 320;
constexpr int   VD = 256, VH = 256, VW = 256;
constexpr float MIN_DEPTH    = 3.0f;
constexpr float MAX_DEPTH    = 9.0f;
constexpr float SCALING      = 0.1f;
constexpr float TAN_HALF_FOV = 1.0f / 3.0f;
constexpr float EPS_EA       = 1e-10f;
constexpr float EPS_MIN      = 1e-8f;
constexpr float HALF_EXT     = 255.0f / 256.0f;   // voxel_size*(n-1)/2
constexpr int   PREFETCH_AHEAD = 16;
}

__device__ __forceinline__ int clampi(int v, int lo, int hi) {
  return v < lo ? lo : (v > hi ? hi : v);
}

// Low 32 bits of a flat shared-memory address == LDS byte offset.
__device__ __forceinline__ unsigned lds_off(void* p) {
  return (unsigned)(uintptr_t)(void __attribute__((address_space(3)))*)p;
}

// One async 16B-per-lane copy: wave `wave` stages its 512B chunk of the
// 4-row (4 KB) slab for one sample into LDS.  rows: 0:(z0,y0) 1:(z0,y1)
// 2:(z1,y0) 3:(z1,y1); each row is a full 256-float volume x-line.
__device__ __forceinline__ void stage_slab(
    const float* __restrict__ vol, float* __restrict__ dst,
    int oy0, int oy1, int oz0, int oz1, int wave, int lane)
{
  const int r = wave >> 1;                 // row 0..3
  const int h = wave & 1;                  // half 0..1
  int rowbase;
  switch (r) {
    case 0:  rowbase = oz0 + oy0; break;
    case 1:  rowbase = oz0 + oy1; break;
    case 2:  rowbase = oz1 + oy0; break;
    default: rowbase = oz1 + oy1; break;
  }
  const int fo = h * 128 + lane * 4;       // float offset within the row
  const float* src = vol + rowbase + fo;   // 16B aligned (rowbase % 256 == 0)
  const unsigned dst_lds = lds_off(dst + r * 256 + fo);
  asm volatile("global_load_async_to_lds_b128 %0, %1, off"
               :: "v"(dst_lds), "v"(src) : "memory");
}

__device__ __forceinline__ void wait_async_and_barrier() {
  asm volatile("s_wait_asynccnt 0x0" ::: "memory");
  __syncthreads();
}

__global__ void zero_max_kernel(unsigned int* maxbits) {
  if (threadIdx.x == 0) *maxbits = 0u;
}

__global__ __launch_bounds__(256) void xray_render_kernel(
    const float* __restrict__ vol,    // [256][256][256] (z,y,x)
    const float* __restrict__ rot,    // [3][3] row-major
    const float* __restrict__ cpos,   // [3]
    float* __restrict__ img_t,        // d_out, transposed layout [x][y]
    unsigned int* __restrict__ maxbits)
{
  const int p  = blockIdx.x * 256 + threadIdx.x;   // one block == one image row
  const int px = p & (IMG_W - 1);
  const int py = p >> 8;

  const float gx = -1.0f + (2.0f / (IMG_W - 1)) * (float)px;
  const float gy = -1.0f + (2.0f / (IMG_H - 1)) * (float)py;
  const float dcx = gx * TAN_HALF_FOV;
  const float dcy = gy * TAN_HALF_FOV;

  const float r10 = rot[3], r20 = rot[6];
  // dir_world = R * [dcx, dcy, 1]
  const float dwx = fmaf(rot[0], dcx, fmaf(rot[1], dcy, rot[2]));
  const float dwy = fmaf(r10,    dcx, fmaf(rot[4], dcy, rot[5]));
  const float dwz = fmaf(r20,    dcx, fmaf(rot[7], dcy, rot[8]));
  const float cx = cpos[0], cy = cpos[1], cz = cpos[2];

  // voxel coord is affine in depth: f = a + b*d
  const float sx = 0.5f * (float)(VW - 1) / HALF_EXT;   // == 128
  const float sy = 0.5f * (float)(VH - 1) / HALF_EXT;
  const float sz = 0.5f * (float)(VD - 1) / HALF_EXT;
  const float ax = fmaf(cx, sx, 0.5f * (float)(VW - 1));
  const float ay = fmaf(cy, sy, 0.5f * (float)(VH - 1));
  const float az = fmaf(cz, sz, 0.5f * (float)(VD - 1));
  const float bx = dwx * sx, by = dwy * sy, bz = dwz * sz;

  const float dstep = (MAX_DEPTH - MIN_DEPTH) / (float)(N_PTS - 1);

  float A   = 1.0f;   // running prod(1 + EPS_EA - dens)
  float P   = 1.0f;   // running prod(1 - dens)
  float rgb = 0.0f;

  __shared__ __align__(16) float slab[2][4 * 256];  // double-buffered 4-row slab
  __shared__ float smax[8];

  const int wave = threadIdx.x >> 5;
  const int lane = threadIdx.x & 31;

  // fy/fz are block-uniform per sample iff dwy,dwz don't depend on dcx.
  // rot[] is identical for every thread -> this branch is block-uniform,
  // so the barriers inside the fast path are safe.
  const bool fast = (r10 == 0.0f) && (r20 == 0.0f);

  if (fast) {
    // ----- FAST PATH: async-staged LDS slabs, software-pipelined -----
    {
      const float d0 = MIN_DEPTH;
      const int y0 = (int)floorf(fmaf(by, d0, ay));
      const int z0 = (int)floorf(fmaf(bz, d0, az));
      stage_slab(vol, slab[0],
                 clampi(y0, 0, VH - 1) << 8, clampi(y0 + 1, 0, VH - 1) << 8,
                 clampi(z0, 0, VD - 1) << 16, clampi(z0 + 1, 0, VD - 1) << 16,
                 wave, lane);
      wait_async_and_barrier();
    }

    for (int s = 0; s < N_PTS; ++s) {
      // kick off DMA for sample s+1 into the other buffer
      if (s + 1 < N_PTS) {
        const float d2 = fmaf(dstep, (float)(s + 1), MIN_DEPTH);
        const int ny0 = (int)floorf(fmaf(by, d2, ay));
        const int nz0 = (int)floorf(fmaf(bz, d2, az));
        stage_slab(vol, slab[(s + 1) & 1],
                   clampi(ny0, 0, VH - 1) << 8, clampi(ny0 + 1, 0, VH - 1) << 8,
                   clampi(nz0, 0, VD - 1) << 16, clampi(nz0 + 1, 0, VD - 1) << 16,
                   wave, lane);
      }

      // compute sample s from the staged slab
      const float d  = fmaf(dstep, (float)s, MIN_DEPTH);
      const float fx = fmaf(bx, d, ax);
      const float fy = fmaf(by, d, ay);   // uniform across block
      const float fz = fmaf(bz, d, az);   // uniform across block

      const int x0 = (int)floorf(fx), x1 = x0 + 1;
      const int y0 = (int)floorf(fy), y1 = y0 + 1;
      const int z0 = (int)floorf(fz), z1 = z0 + 1;
      const float wx = fx - (float)x0;
      const float wy = fy - (float)y0;
      const float wz = fz - (float)z0;

      const float mx0 = (x0 >= 0 && x0 < VW) ? 1.0f : 0.0f;
      const float mx1 = (x1 >= 0 && x1 < VW) ? 1.0f : 0.0f;
      const float my0 = (y0 >= 0 && y0 < VH) ? 1.0f : 0.0f;
      const float my1 = (y1 >= 0 && y1 < VH) ? 1.0f : 0.0f;
      const float mz0 = (z0 >= 0 && z0 < VD) ? 1.0f : 0.0f;
      const float mz1 = (z1 >= 0 && z1 < VD) ? 1.0f : 0.0f;

      const int ox0 = clampi(x0, 0, VW - 1);
      const int ox1 = clampi(x1, 0, VW - 1);

      const float* R0 = slab[s & 1];
      const float m00 = mz0 * my0, m01 = mz0 * my1;
      const float m10 = mz1 * my0, m11 = mz1 * my1;
      const float c000 = R0[0 * 256 + ox0] * (m00 * mx0);
      const float c001 = R0[0 * 256 + ox1] * (m00 * mx1);
      const float c010 = R0[1 * 256 + ox0] * (m01 * mx0);
      const float c011 = R0[1 * 256 + ox1] * (m01 * mx1);
      const float c100 = R0[2 * 256 + ox0] * (m10 * mx0);
      const float c101 = R0[2 * 256 + ox1] * (m10 * mx1);
      const float c110 = R0[3 * 256 + ox0] * (m11 * mx0);
      const float c111 = R0[3 * 256 + ox1] * (m11 * mx1);

      const float f00 = fmaf(wx, c001 - c000, c000);
      const float f01 = fmaf(wx, c011 - c010, c010);
      const float f10 = fmaf(wx, c101 - c100, c100);
      const float f11 = fmaf(wx, c111 - c110, c110);
      const float f0  = fmaf(wy, f01 - f00, f00);
      const float f1  = fmaf(wy, f11 - f10, f10);
      const float feat = fmaf(wz, f1 - f0, f0);

      const float covx = fmaf(1.0f - wx, mx0, wx * mx1);
      const float covy = fmaf(1.0f - wy, my0, wy * my1);
      const float covz = fmaf(1.0f - wz, mz0, wz * mz1);
      const float dens = SCALING * covx * covy * covz;

      rgb = fmaf(dens * A, feat, rgb);
      A  *= (1.0f + EPS_EA - dens);
      P  *= (1.0f - dens);

      // drain own async op, then block barrier -> next buffer fully staged
      wait_async_and_barrier();
    }
  } else {
    // ----- GENERIC PATH: branch-free masked gathers + prefetch -----
    #pragma unroll 4
    for (int s = 0; s < N_PTS; ++s) {
      const float d  = fmaf(dstep, (float)s, MIN_DEPTH);
      const float fx = fmaf(bx, d, ax);
      const float fy = fmaf(by, d, ay);
      const float fz = fmaf(bz, d, az);

      if ((s & 3) == 0 && (s + PREFETCH_AHEAD) < N_PTS) {
        const float d2 = fmaf(dstep, (float)(s + PREFETCH_AHEAD), MIN_DEPTH);
        const int pxi = clampi((int)floorf(fmaf(bx, d2, ax)), 0, VW - 1);
        const int pyi = clampi((int)floorf(fmaf(by, d2, ay)), 0, VH - 1);
        const int pzi = clampi((int)floorf(fmaf(bz, d2, az)), 0, VD - 1);
        __builtin_prefetch(vol + ((pzi << 16) | (pyi << 8) | pxi), 0, 3);
      }

      const int x0 = (int)floorf(fx), x1 = x0 + 1;
      const int y0 = (int)floorf(fy), y1 = y0 + 1;
      const int z0 = (int)floorf(fz), z1 = z0 + 1;
      const float wx = fx - (float)x0;
      const float wy = fy - (float)y0;
      const float wz = fz - (float)z0;

      const float mx0 = (x0 >= 0 && x0 < VW) ? 1.0f : 0.0f;
      const float mx1 = (x1 >= 0 && x1 < VW) ? 1.0f : 0.0f;
      const float my0 = (y0 >= 0 && y0 < VH) ? 1.0f : 0.0f;
      const float my1 = (y1 >= 0 && y1 < VH) ? 1.0f : 0.0f;
      const float mz0 = (z0 >= 0 && z0 < VD) ? 1.0f : 0.0f;
      const float mz1 = (z1 >= 0 && z1 < VD) ? 1.0f : 0.0f;

      const int ox0 = clampi(x0, 0, VW - 1);
      const int ox1 = clampi(x1, 0, VW - 1);
      const int oy0 = clampi(y0, 0, VH - 1) << 8;
      const int oy1 = clampi(y1, 0, VH - 1) << 8;
      const int oz0 = clampi(z0, 0, VD - 1) << 16;
      const int oz1 = clampi(z1, 0, VD - 1) << 16;

      const float m00 = mz0 * my0, m01 = mz0 * my1;
      const float m10 = mz1 * my0, m11 = mz1 * my1;
      const float c000 = vol[oz0 + oy0 + ox0] * (m00 * mx0);
      const float c001 = vol[oz0 + oy0 + ox1] * (m00 * mx1);
      const float c010 = vol[oz0 + oy1 + ox0] * (m01 * mx0);
      const float c011 = vol[oz0 + oy1 + ox1] * (m01 * mx1);
      const float c100 = vol[oz1 + oy0 + ox0] * (m10 * mx0);
      const float c101 = vol[oz1 + oy0 + ox1] * (m10 * mx1);
      const float c110 = vol[oz1 + oy1 + ox0] * (m11 * mx0);
      const float c111 = vol[oz1 + oy1 + ox1] * (m11 * mx1);

      const float f00 = fmaf(wx, c001 - c000, c000);
      const float f01 = fmaf(wx, c011 - c010, c010);
      const float f10 = fmaf(wx, c101 - c100, c100);
      const float f11 = fmaf(wx, c111 - c110, c110);
      const float f0  = fmaf(wy, f01 - f00, f00);
      const float f1  = fmaf(wy, f11 - f10, f10);
      const float feat = fmaf(wz, f1 - f0, f0);

      const float covx = fmaf(1.0f - wx, mx0, wx * mx1);
      const float covy = fmaf(1.0f - wy, my0, wy * my1);
      const float covz = fmaf(1.0f - wz, mz0, wz * mz1);
      const float dens = SCALING * covx * covy * covz;

      rgb = fmaf(dens * A, feat, rgb);
      A  *= (1.0f + EPS_EA - dens);
      P  *= (1.0f - dens);
    }
  }

  const float gray = (3.0f * rgb + (1.0f - P)) * 0.25f;

  // transposed output layout: out[0,0,x,y]
  img_t[px * IMG_H + py] = gray;

  // block max -> global atomic max (gray >= 0 -> uint-bit max exact)
  float m = gray;
  #pragma unroll
  for (int off = 16; off > 0; off >>= 1)
    m = fmaxf(m, __shfl_down(m, off, 32));
  if (lane == 0) smax[wave] = m;
  __syncthreads();
  if (wave == 0) {
    m = (lane < 8) ? smax[lane] : 0.0f;
    #pragma unroll
    for (int off = 4; off > 0; off >>= 1)
      m = fmaxf(m, __shfl_down(m, off, 32));
    if (lane == 0) atomicMax(maxbits, __float_as_uint(m));
  }
}

__global__ __launch_bounds__(256) void normalize_kernel(
    float* __restrict__ img, const unsigned int* __restrict__ maxbits)
{
  const int q = blockIdx.x * 256 + threadIdx.x;
  const float mv = __uint_as_float(*maxbits);
  img[q] = (img[q] + EPS_MIN) / (mv + EPS_MIN);
}

extern "C" void kernel_launch(void* const* d_in, const int* in_sizes, int n_in,
                              void* d_out, int out_size, void* d_ws, size_t ws_size,
                              hipStream_t stream) {
  (void)in_sizes; (void)n_in; (void)out_size; (void)ws_size;
  const float* vol = (const float*)d_in[0];   // image3d (1,1,256,256,256) f32
  const float* rot = (const float*)d_in[1];   // cam_rot (1,3,3) f32
  const float* pos = (const float*)d_in[2];   // cam_pos (1,3) f32
  float* out = (float*)d_out;                 // (1,1,256,256) f32
  unsigned int* maxbits = (unsigned int*)d_ws;

  zero_max_kernel<<<1, 32, 0, stream>>>(maxbits);
  xray_render_kernel<<<(IMG_W * IMG_H) / 256, 256, 0, stream>>>(
      vol, rot, pos, out, maxbits);
  normalize_kernel<<<(IMG_W * IMG_H) / 256, 256, 0, stream>>>(out, maxbits);
}